// RAMGen_21363167330524
// MI455X (gfx1250) — compile-verified
//
#include <hip/hip_runtime.h>
#include <math.h>
#include <stdint.h>

// ---------------------------------------------------------------------------
// RAMGen for MI455X (gfx1250). f32 end-to-end; matmuls on V_WMMA_F32_16X16X4_F32
// (bandwidth-bound problem -> keep reference f32 numerics). LDS staging of
// pure copies uses GLOBAL_LOAD_ASYNC_TO_LDS_B32 (ASYNCcnt) when available.
// ---------------------------------------------------------------------------

#define EPSF 1e-5f

typedef float v2f __attribute__((ext_vector_type(2)));
typedef float v8f __attribute__((ext_vector_type(8)));

// ---- gfx1250 async global->LDS copy (no VGPR round trip) -------------------
#if defined(__AMDGCN__) && defined(__has_builtin)
#if __has_builtin(__builtin_amdgcn_global_load_async_to_lds_b32)
#define HAVE_ASYNC_LDS 1
#endif
#endif
#ifndef HAVE_ASYNC_LDS
#define HAVE_ASYNC_LDS 0
#endif

__device__ __forceinline__ void async_copy_b32(const float* g, float* l) {
#if HAVE_ASYNC_LDS
  __builtin_amdgcn_global_load_async_to_lds_b32(
      (__attribute__((address_space(1))) int*)(g),
      (__attribute__((address_space(3))) int*)(l), 0, 0);
#else
  *l = *g;
#endif
}

__device__ __forceinline__ void wait_async0() {
#if HAVE_ASYNC_LDS
#if __has_builtin(__builtin_amdgcn_s_wait_asynccnt)
  __builtin_amdgcn_s_wait_asynccnt(0);
#else
  asm volatile("s_wait_asynccnt 0" ::: "memory");
#endif
#endif
}

// D(16x16,f32) = A(16x4,f32) * B(4x16,f32) + C
// A layout: lane<16 holds M=lane,K={0,1}; lane>=16 holds M=lane-16,K={2,3}
// B layout: lane<16 holds N=lane,K={0,1}; lane>=16 holds N=lane-16,K={2,3}
// C/D: VGPR j -> row M=j (lanes 0-15) / M=j+8 (lanes 16-31), col N=lane&15
__device__ __forceinline__ v8f wmma4(v2f a, v2f b, v8f c) {
  return __builtin_amdgcn_wmma_f32_16x16x4_f32(false, a, false, b, (short)0, c,
                                               false, false);
}

// ---------------------------------------------------------------------------
// Fused spatial graph conv: out[n,cc,t,w] = sum_k sum_v (W_k x + b_k)[cc,v] A_k[v,w]
// One block per (n,t). 256 threads = 8 waves; 8 output tiles (4 M x 2 N).
// BN_IN: apply BN(mean/rstd/gamma/beta per input channel) + ReLU to input.
// ---------------------------------------------------------------------------
template <int CIN, bool BN_IN>
__global__ void __launch_bounds__(256)
gc_kernel(const float* __restrict__ xg, int m, const float* __restrict__ W,
          const float* __restrict__ bias, const float* __restrict__ Ag,
          const float* __restrict__ bn_mean, const float* __restrict__ bn_rstd,
          const float* __restrict__ bn_gamma, const float* __restrict__ bn_beta,
          float* __restrict__ outg) {
  constexpr int CINP = (CIN + 3) & ~3;       // K padded to multiple of 4
  constexpr int KST1 = CINP / 4;             // K-steps for channel GEMM

  __shared__ float xin_s[64 * 33];           // [CINP][32] input, zero padded
  __shared__ float y_s[64 * 33];             // [64][32] per-k y tile
  __shared__ float amat_s[3 * 28 * 33];      // [k][28][32] A, zero padded

  const int nt = blockIdx.x;
  const int n = nt / 300, t = nt - n * 300;
  const int tid = threadIdx.x;
  const int lane = tid & 31, wave = tid >> 5;
  const int l15 = lane & 15, hi = lane >> 4;
  const int tileM = wave >> 1, tileN = wave & 1;

  for (int i = tid; i < CINP * 33; i += 256) xin_s[i] = 0.f;
  for (int i = tid; i < 3 * 28 * 33; i += 256) amat_s[i] = 0.f;
  __syncthreads();
  // adjacency: pure copy -> async DMA to LDS (overlaps with input transform)
  for (int i = tid; i < 3 * 625; i += 256) {
    int k = i / 625, r = i - k * 625, v = r / 25, w = r - v * 25;
    async_copy_b32(Ag + i, &amat_s[k * 924 + v * 33 + w]);
  }
  if (BN_IN) {
    // elementwise BN+ReLU transform in flight: needs VGPR path
    for (int i = tid; i < CIN * 25; i += 256) {
      int c = i / 25, v = i - c * 25;
      float val = xg[((size_t)(n * 64 + c) * 300 + t) * 25 + v];
      val = (val - bn_mean[c]) * bn_rstd[c] * bn_gamma[c] + bn_beta[c];
      xin_s[c * 33 + v] = val > 0.f ? val : 0.f;
    }
  } else {
    // raw input slice: pure copy -> async DMA to LDS
    for (int i = tid; i < CIN * 25; i += 256) {
      int c = i / 25, v = i - c * 25;
      async_copy_b32(xg + ((size_t)((n * 2 + m) * 3 + c) * 300 + t) * 25 + v,
                     &xin_s[c * 33 + v]);
    }
  }
  wait_async0();
  __syncthreads();

  v8f eacc = {};                                         // wave's 16x16 e tile
  for (int k = 0; k < 3; ++k) {
    // ---- GEMM1: y_k(64x32) = W_k(64xCIN) * xin(CINx32) ----
    v8f yacc = {};
    for (int k4 = 0; k4 < KST1; ++k4) {
      int kk = k4 * 4 + hi * 2;
      int o = k * 64 + tileM * 16 + l15;
      v2f af, bf;
      af.x = (kk < CIN) ? W[o * CIN + kk] : 0.f;
      af.y = (kk + 1 < CIN) ? W[o * CIN + kk + 1] : 0.f;
      bf.x = xin_s[kk * 33 + tileN * 16 + l15];
      bf.y = xin_s[(kk + 1) * 33 + tileN * 16 + l15];
      yacc = wmma4(af, bf, yacc);
    }
    __syncthreads();   // prior GEMM2 reads of y_s done
    {
      int col = tileN * 16 + l15;
#pragma unroll
      for (int j = 0; j < 8; ++j) {
        int row = tileM * 16 + j + hi * 8;
        float bv = (col < 25) ? bias[k * 64 + row] : 0.f;
        y_s[row * 33 + col] = yacc[j] + bv;
      }
    }
    __syncthreads();
    // ---- GEMM2: e(64x25) += y_k(64x25pad28) * A_k(25pad28 x 25) ----
    for (int k4 = 0; k4 < 7; ++k4) {
      int kk = k4 * 4 + hi * 2;
      v2f af, bf;
      af.x = y_s[(tileM * 16 + l15) * 33 + kk];
      af.y = y_s[(tileM * 16 + l15) * 33 + kk + 1];
      bf.x = amat_s[k * 924 + kk * 33 + tileN * 16 + l15];
      bf.y = amat_s[k * 924 + (kk + 1) * 33 + tileN * 16 + l15];
      eacc = wmma4(af, bf, eacc);
    }
  }

  {
    int w = tileN * 16 + l15;
    if (w < 25) {
#pragma unroll
      for (int j = 0; j < 8; ++j) {
        int cc = tileM * 16 + j + hi * 8;
        outg[((size_t)(n * 64 + cc) * 300 + t) * 25 + w] = eacc[j];
      }
    }
  }
}

// ---------------------------------------------------------------------------
// BatchNorm stats, split reduction. Stage 1: one block per (ch, nb) reduces
// L contiguous floats (float4 when aligned) -> partial sum/sumsq.
// Stage 2: one block per channel combines NB partials -> mean, rsqrt(var+eps).
// ---------------------------------------------------------------------------
__global__ void __launch_bounds__(256)
bn_part_kernel(const float* __restrict__ buf, float* __restrict__ part,
               int CH, int L) {
  int ch = blockIdx.x, nb = blockIdx.y;
  const float* p = buf + (size_t)(nb * CH + ch) * L;
  float s = 0.f, s2 = 0.f;
  if (((L & 3) == 0) && ((((uintptr_t)p) & 15) == 0)) {
    const float4* p4 = (const float4*)p;
    int L4 = L >> 2;
    for (int i = threadIdx.x; i < L4; i += 256) {
      float4 v = p4[i];
      s += v.x + v.y + v.z + v.w;
      s2 += v.x * v.x + v.y * v.y + v.z * v.z + v.w * v.w;
    }
  } else {
    for (int i = threadIdx.x; i < L; i += 256) {
      float v = p[i];
      s += v;
      s2 += v * v;
    }
  }
  __shared__ float rs[256], rq[256];
  rs[threadIdx.x] = s;
  rq[threadIdx.x] = s2;
  __syncthreads();
  for (int off = 128; off > 0; off >>= 1) {
    if (threadIdx.x < off) {
      rs[threadIdx.x] += rs[threadIdx.x + off];
      rq[threadIdx.x] += rq[threadIdx.x + off];
    }
    __syncthreads();
  }
  if (threadIdx.x == 0) {
    int nblk = gridDim.y;
    part[(ch * nblk + nb) * 2 + 0] = rs[0];
    part[(ch * nblk + nb) * 2 + 1] = rq[0];
  }
}

__global__ void __launch_bounds__(32)
bn_fin_kernel(const float* __restrict__ part, float* __restrict__ mean_o,
              float* __restrict__ rstd_o, int NB, int L) {
  int ch = blockIdx.x;
  if (threadIdx.x == 0) {
    float s = 0.f, s2 = 0.f;
    for (int nb = 0; nb < NB; ++nb) {
      s += part[(ch * NB + nb) * 2 + 0];
      s2 += part[(ch * NB + nb) * 2 + 1];
    }
    float inv = 1.f / ((float)NB * (float)L);
    float mu = s * inv;
    float var = s2 * inv - mu * mu;
    mean_o[ch] = mu;
    rstd_o[ch] = rsqrtf(var + EPSF);
  }
}

// ---------------------------------------------------------------------------
// RAM_r[n,t,v,w] = sum_c BN(d1)[n,c,t,v] * BN(d2)[n,c,t,w]
// One block per (n,t); 128 threads = 4 waves = 2x2 16x16 tiles, K=64.
// ---------------------------------------------------------------------------
__global__ void __launch_bounds__(128)
ramr_kernel(const float* __restrict__ d1, const float* __restrict__ d2,
            const float* __restrict__ m1, const float* __restrict__ r1,
            const float* __restrict__ m2, const float* __restrict__ r2,
            const float* __restrict__ g, const float* __restrict__ b,
            float* __restrict__ outg) {
  __shared__ float a_s[32 * 65];   // d1 transposed: [v(pad32)][c=64]
  __shared__ float b_s[64 * 33];   // d2: [c=64][w(pad32)]
  int nt = blockIdx.x;
  int n = nt / 300, t = nt - n * 300;
  int tid = threadIdx.x;
  for (int i = tid; i < 32 * 65; i += 128) a_s[i] = 0.f;
  for (int i = tid; i < 64 * 33; i += 128) b_s[i] = 0.f;
  __syncthreads();
  for (int i = tid; i < 64 * 25; i += 128) {
    int c = i / 25, v = i - c * 25;
    size_t idx = ((size_t)(n * 64 + c) * 300 + t) * 25 + v;
    a_s[v * 65 + c] = (d1[idx] - m1[c]) * r1[c] * g[c] + b[c];
    b_s[c * 33 + v] = (d2[idx] - m2[c]) * r2[c] * g[c] + b[c];
  }
  __syncthreads();
  int lane = tid & 31, wave = tid >> 5;
  int l15 = lane & 15, hi = lane >> 4;
  int tileM = wave >> 1, tileN = wave & 1;
  v8f acc = {};
  for (int k4 = 0; k4 < 16; ++k4) {
    int kk = k4 * 4 + hi * 2;
    v2f af, bf;
    af.x = a_s[(tileM * 16 + l15) * 65 + kk];
    af.y = a_s[(tileM * 16 + l15) * 65 + kk + 1];
    bf.x = b_s[kk * 33 + tileN * 16 + l15];
    bf.y = b_s[(kk + 1) * 33 + tileN * 16 + l15];
    acc = wmma4(af, bf, acc);
  }
  int w = tileN * 16 + l15;
  if (w < 25) {
#pragma unroll
    for (int j = 0; j < 8; ++j) {
      int v = tileM * 16 + j + hi * 8;
      if (v < 25) outg[(size_t)nt * 625 + v * 25 + w] = acc[j];
    }
  }
}

// ---------------------------------------------------------------------------
// Final: RAM_g from x, BN(T)+scaled_tanh on RAM_r, integrate, mask,
// threshold, degree-normalize. One block per (n,t).
// ---------------------------------------------------------------------------
__global__ void __launch_bounds__(256)
final_kernel(const float* __restrict__ xg, const float* __restrict__ ramr,
             const float* __restrict__ ram_mean, const float* __restrict__ ram_rstd,
             const float* __restrict__ ram_gamma, const float* __restrict__ ram_beta,
             const float* __restrict__ ip, float* __restrict__ outg) {
  __shared__ float ea_s[75], eb_s[75], xb_s[75];
  __shared__ float var_s[3];
  __shared__ float ram_s[25 * 26];
  __shared__ float da_s[25], db_s[25];
  int nt = blockIdx.x;
  int n = nt / 300, t = nt - n * 300;
  int tid = threadIdx.x;
  if (tid < 75) {
    int c = tid / 25, v = tid - c * 25;
    float xa = xg[((size_t)((n * 2 + 0) * 3 + c) * 300 + t) * 25 + v];
    float xb = xg[((size_t)((n * 2 + 1) * 3 + c) * 300 + t) * 25 + v];
    ea_s[tid] = expf(xa);
    eb_s[tid] = expf(-xb);
    xb_s[tid] = xb;
  }
  __syncthreads();
  if (tid < 3) {   // var over V (ddof=1) of xb, per channel
    float s = 0.f;
    for (int v = 0; v < 25; ++v) s += xb_s[tid * 25 + v];
    float mu = s * (1.f / 25.f);
    float ss = 0.f;
    for (int v = 0; v < 25; ++v) {
      float d = xb_s[tid * 25 + v] - mu;
      ss += d * d;
    }
    var_s[tid] = ss * (1.f / 24.f);
  }
  __syncthreads();
  float mask = (var_s[0] + var_s[1] + var_s[2] < 1e-4f) ? 0.f : 1.f;
  float p = (tanhf(ip[0]) + 1.f) * 0.5f;
  float gm = ram_gamma[t], bt = ram_beta[t];
  float mu = ram_mean[t], rs = ram_rstd[t];
  for (int i = tid; i < 625; i += 256) {
    int v = i / 25, w = i - v * 25;
    float gsum = 0.f;
#pragma unroll
    for (int c = 0; c < 3; ++c) {
      float l = logf(ea_s[c * 25 + v] * eb_s[c * 25 + w] + EPSF);
      gsum += l * l;
    }
    float rg = expf(-gsum * (1.f / 3.f));
    float rr = ramr[(size_t)nt * 625 + i];
    float rt = (tanhf((rr - mu) * rs * gm + bt) + 1.f) * 0.5f;
    float val = p * rt + (1.f - p) * rg;
    val *= mask;
    val = (val < 0.5f) ? 0.f : val;
    ram_s[v * 26 + w] = val;
  }
  __syncthreads();
  if (tid < 25) {                      // da[w] = sum_v RAM
    float s = 0.f;
    for (int v = 0; v < 25; ++v) s += ram_s[v * 26 + tid];
    da_s[tid] = s;
  } else if (tid >= 32 && tid < 57) {  // db[v] = sum_w RAM
    int v = tid - 32;
    float s = 0.f;
    for (int w = 0; w < 25; ++w) s += ram_s[v * 26 + w];
    db_s[v] = s;
  }
  __syncthreads();
  for (int i = tid; i < 625; i += 256) {
    int v = i / 25, w = i - v * 25;
    outg[(size_t)nt * 625 + i] = ram_s[v * 26 + w] * rsqrtf(db_s[v] * da_s[w] + EPSF);
  }
}

// ---------------------------------------------------------------------------
extern "C" void kernel_launch(void* const* d_in, const int* in_sizes, int n_in,
                              void* d_out, int out_size, void* d_ws,
                              size_t ws_size, hipStream_t stream) {
  const float* x    = (const float*)d_in[0];
  const float* A    = (const float*)d_in[1];
  const float* encW = (const float*)d_in[2];
  const float* encb = (const float*)d_in[3];
  const float* decW = (const float*)d_in[4];
  const float* decb = (const float*)d_in[5];
  const float* geg  = (const float*)d_in[6];   // bn_enc_gamma
  const float* geb  = (const float*)d_in[7];   // bn_enc_beta
  const float* gdg  = (const float*)d_in[8];   // bn_dec_gamma
  const float* gdb  = (const float*)d_in[9];   // bn_dec_beta
  const float* grg  = (const float*)d_in[10];  // bn_ram_gamma
  const float* grb  = (const float*)d_in[11];  // bn_ram_beta
  const float* ip   = (const float*)d_in[12];  // integration_param

  float* out = (float*)d_out;
  float* ws = (float*)d_ws;
  const size_t E = (size_t)32 * 64 * 300 * 25;  // 15.36M floats per activation
  float* e1 = ws;
  float* e2 = ws + E;
  float* dd1 = ws + 2 * E;
  float* dd2 = ws + 3 * E;
  float* ramr = ws;           // reuse e1 region (dead after dec pass)
  float* st = ws + 4 * E;     // stats area
  float *e1m = st, *e1r = st + 64, *e2m = st + 128, *e2r = st + 192;
  float *d1m = st + 256, *d1r = st + 320, *d2m = st + 384, *d2r = st + 448;
  float *rm = st + 512, *rrs = st + 812;
  float* part = st + 2048;    // split-reduction partials (<= 300*32*2 floats)

  dim3 g(32 * 300);
  // encoder GC (pre-BN outputs)
  gc_kernel<3, false><<<g, 256, 0, stream>>>(x, 0, encW, encb, A, nullptr,
                                             nullptr, nullptr, nullptr, e1);
  gc_kernel<3, false><<<g, 256, 0, stream>>>(x, 1, encW, encb, A, nullptr,
                                             nullptr, nullptr, nullptr, e2);
  bn_part_kernel<<<dim3(64, 32), 256, 0, stream>>>(e1, part, 64, 7500);
  bn_fin_kernel<<<64, 32, 0, stream>>>(part, e1m, e1r, 32, 7500);
  bn_part_kernel<<<dim3(64, 32), 256, 0, stream>>>(e2, part, 64, 7500);
  bn_fin_kernel<<<64, 32, 0, stream>>>(part, e2m, e2r, 32, 7500);
  // decoder GC with fused BN+ReLU on input
  gc_kernel<64, true><<<g, 256, 0, stream>>>(e1, 0, decW, decb, A, e1m, e1r,
                                             geg, geb, dd1);
  gc_kernel<64, true><<<g, 256, 0, stream>>>(e2, 0, decW, decb, A, e2m, e2r,
                                             geg, geb, dd2);
  bn_part_kernel<<<dim3(64, 32), 256, 0, stream>>>(dd1, part, 64, 7500);
  bn_fin_kernel<<<64, 32, 0, stream>>>(part, d1m, d1r, 32, 7500);
  bn_part_kernel<<<dim3(64, 32), 256, 0, stream>>>(dd2, part, 64, 7500);
  bn_fin_kernel<<<64, 32, 0, stream>>>(part, d2m, d2r, 32, 7500);
  // RAM_r einsum with fused BN on both operands
  ramr_kernel<<<g, 128, 0, stream>>>(dd1, dd2, d1m, d1r, d2m, d2r, gdg, gdb,
                                     ramr);
  bn_part_kernel<<<dim3(300, 32), 256, 0, stream>>>(ramr, part, 300, 625);
  bn_fin_kernel<<<300, 32, 0, stream>>>(part, rm, rrs, 32, 625);
  // final fusion: RAM_g + BN(T)/tanh + mask + threshold + normalize
  final_kernel<<<g, 256, 0, stream>>>(x, ramr, rm, rrs, grg, grb, ip, out);
}